// GNNpool_24696061952388
// MI455X (gfx1250) — compile-verified
//
#include <hip/hip_runtime.h>
#include <math.h>
#include <stdint.h>

typedef __attribute__((ext_vector_type(2))) float v2f;
typedef __attribute__((ext_vector_type(8))) float v8f;
typedef unsigned long long u64;

// ---------------- elementwise helpers ----------------
__global__ void fill_f32(float* __restrict__ p, float v, int n) {
  int i = blockIdx.x * blockDim.x + threadIdx.x;
  if (i < n) p[i] = v;
}
__global__ void fill_i32(int* __restrict__ p, int v, int n) {
  int i = blockIdx.x * blockDim.x + threadIdx.x;
  if (i < n) p[i] = v;
}

// ---------------- WMMA fp32 GEMM: C[M,N] = A[M,K] @ B[K,N] ----------------
// One wave computes a 16x64 tile: 4 accumulators share each A fragment (4x A
// reuse, 4 WMMAs per load batch keeps the XDL pipe fed between waitcnts).
__global__ void gemm_wmma(const float* __restrict__ A, const float* __restrict__ B,
                          float* __restrict__ C, int M, int N, int K, int totalWaves) {
  int gw = blockIdx.x * (blockDim.x >> 5) + (threadIdx.x >> 5);
  if (gw >= totalWaves) return;                 // wave-uniform: EXEC stays all-ones
  int tilesN4 = N >> 6;
  int tm = gw / tilesN4;
  int tn = gw - tm * tilesN4;

  int lane = threadIdx.x & 31;
  int half = lane >> 4;                          // 0: K pair {0,1}, 1: K pair {2,3}
  int l16  = lane & 15;

  const float* Arow  = A + (size_t)(tm * 16 + l16) * K + 2 * half;
  int col = tn * 64 + l16;
  const float* Bbase = B + (size_t)(2 * half) * N + col;

  v8f acc0 = {}, acc1 = {}, acc2 = {}, acc3 = {};
  for (int k = 0; k < K; k += 4) {
    v2f a = *(const v2f*)(Arow + k);             // A[m, k+2h .. k+2h+1]
    const float* bk = Bbase + (size_t)k * N;
    v2f b0 = { bk[0],  bk[N] };                  // B[k+2h.., col]
    v2f b1 = { bk[16], bk[16 + N] };
    v2f b2 = { bk[32], bk[32 + N] };
    v2f b3 = { bk[48], bk[48 + N] };
    acc0 = __builtin_amdgcn_wmma_f32_16x16x4_f32(false, a, false, b0, (short)0, acc0, false, false);
    acc1 = __builtin_amdgcn_wmma_f32_16x16x4_f32(false, a, false, b1, (short)0, acc1, false, false);
    acc2 = __builtin_amdgcn_wmma_f32_16x16x4_f32(false, a, false, b2, (short)0, acc2, false, false);
    acc3 = __builtin_amdgcn_wmma_f32_16x16x4_f32(false, a, false, b3, (short)0, acc3, false, false);
  }
  float* Cb = C + (size_t)(tm * 16 + 8 * half) * N + col;
#pragma unroll
  for (int r = 0; r < 8; ++r) {
    Cb[(size_t)r * N]      = acc0[r];
    Cb[(size_t)r * N + 16] = acc1[r];
    Cb[(size_t)r * N + 32] = acc2[r];
    Cb[(size_t)r * N + 48] = acc3[r];
  }
}

// ---------------- graph normalization ----------------
__global__ void deg_accum(const int* __restrict__ dst, const float* __restrict__ emask,
                          float* __restrict__ deg, int E) {
  int e = blockIdx.x * blockDim.x + threadIdx.x;
  if (e >= E) return;
  atomicAdd(&deg[dst[e]], emask ? emask[e] : 1.0f);
}
__global__ void edge_norm(const int* __restrict__ src, const int* __restrict__ dst,
                          const float* __restrict__ emask, const float* __restrict__ deg,
                          float* __restrict__ nrm, int E) {
  int e = blockIdx.x * blockDim.x + threadIdx.x;
  if (e >= E) return;
  float m = emask ? emask[e] : 1.0f;
  nrm[e] = m * rsqrtf(deg[src[e]] * deg[dst[e]]);
}

// agg[dst] += h[src] * norm  (4 features per thread; cs = log2(F/4))
__global__ void agg_edges(const int* __restrict__ src, const int* __restrict__ dst,
                          const float* __restrict__ nrm, const float* __restrict__ h,
                          float* __restrict__ agg, int E, int F, int cs) {
  int t = blockIdx.x * blockDim.x + threadIdx.x;
  int e = t >> cs;
  if (e >= E) return;
  float w = nrm[e];
  if (w == 0.0f) return;
  int f = (t & ((1 << cs) - 1)) << 2;
  const float* hp = h + (size_t)src[e] * F + f;
  float*       ap = agg + (size_t)dst[e] * F + f;
  float4 hv = *(const float4*)hp;
  atomicAdd(ap + 0, hv.x * w);
  atomicAdd(ap + 1, hv.y * w);
  atomicAdd(ap + 2, hv.z * w);
  atomicAdd(ap + 3, hv.w * w);
}

// out = relu(agg + h/deg + bias)   (in-place on agg is safe); fs = log2(F)
__global__ void gcn_finalize(const float* __restrict__ agg, const float* __restrict__ h,
                             const float* __restrict__ deg, const float* __restrict__ bias,
                             float* __restrict__ out, int n, int F, int fs) {
  int t = blockIdx.x * blockDim.x + threadIdx.x;
  if (t >= n * F) return;
  int i = t >> fs;
  int f = t & (F - 1);
  float v = agg[t] + h[t] / deg[i] + bias[f];
  out[t] = v > 0.0f ? v : 0.0f;
}

// ---------------- TopK pooling ----------------
__global__ void pnorm_kernel(const float* __restrict__ p, int len, float* __restrict__ pn) {
  __shared__ float red[256];
  int tid = threadIdx.x;
  float s = 0.0f;
  for (int i = tid; i < len; i += 256) s += p[i] * p[i];
  red[tid] = s;
  __syncthreads();
  for (int off = 128; off > 0; off >>= 1) {
    if (tid < off) red[tid] += red[tid + off];
    __syncthreads();
  }
  if (tid == 0) pn[0] = sqrtf(red[0]);
}

// one wave per row: score[i] = dot(x[i], p) / ||p||
__global__ void score_kernel(const float* __restrict__ x, const float* __restrict__ p,
                             const float* __restrict__ pn, float* __restrict__ score,
                             int n, int F) {
  int w = (blockIdx.x * blockDim.x + threadIdx.x) >> 5;
  int lane = threadIdx.x & 31;
  if (w >= n) return;
  const float* row = x + (size_t)w * F;
  float s = 0.0f;
  for (int f = lane; f < F; f += 32) s += row[f] * p[f];
  for (int off = 16; off > 0; off >>= 1) s += __shfl_down(s, off);
  if (lane == 0) score[w] = s / pn[0];
}

// Single-workgroup bitonic sort of up to 32768 u64 keys in 256KB LDS.
// key = (~ord(score) << 32) | index  -> ascending sort == descending score,
// ties broken by ascending index (matches jax.lax.top_k).
__global__ void __launch_bounds__(1024) topk_sort(const float* __restrict__ score,
                                                  int n, int total, int k,
                                                  int* __restrict__ perm,
                                                  float* __restrict__ perm_f) {
  extern __shared__ u64 keys[];
  int tid = threadIdx.x;
  for (int i = tid; i < total; i += blockDim.x) {
    u64 kk;
    if (i < n) {
      unsigned u = __float_as_uint(score[i]);
      unsigned ord = (u & 0x80000000u) ? ~u : (u | 0x80000000u);  // monotone map
      kk = ((u64)(~ord) << 32) | (unsigned)i;
    } else {
      kk = ~0ull;
    }
    keys[i] = kk;
  }
  __syncthreads();
  for (int size = 2; size <= total; size <<= 1) {
    for (int stride = size >> 1; stride > 0; stride >>= 1) {
      for (int i = tid; i < (total >> 1); i += blockDim.x) {
        int pos = 2 * i - (i & (stride - 1));
        u64 a = keys[pos], b = keys[pos + stride];
        bool asc = (pos & size) == 0;
        if ((a > b) == asc) { keys[pos] = b; keys[pos + stride] = a; }
      }
      __syncthreads();
    }
  }
  for (int i = tid; i < k; i += blockDim.x) {
    int idx = (int)(keys[i] & 0xFFFFFFFFu);
    perm[i] = idx;
    perm_f[i] = (float)idx;
  }
}

// xk[i] = x[perm[i]] * tanh(score[perm[i]]);  fs = log2(F)
__global__ void gather_scale(const float* __restrict__ x, const int* __restrict__ perm,
                             const float* __restrict__ score, float* __restrict__ out,
                             int k, int F, int fs) {
  int t = blockIdx.x * blockDim.x + threadIdx.x;
  if (t >= k * F) return;
  int i = t >> fs;
  int f = t & (F - 1);
  int idx = perm[i];
  out[t] = x[(size_t)idx * F + f] * tanhf(score[idx]);
}

__global__ void scatter_map(const int* __restrict__ perm, int* __restrict__ mapping, int k) {
  int i = blockIdx.x * blockDim.x + threadIdx.x;
  if (i < k) mapping[perm[i]] = i;
}

__global__ void relabel(const int* __restrict__ src, const int* __restrict__ dst,
                        const float* __restrict__ emask, const int* __restrict__ mapping,
                        int* __restrict__ nsrc, int* __restrict__ ndst,
                        float* __restrict__ nmask, int E) {
  int e = blockIdx.x * blockDim.x + threadIdx.x;
  if (e >= E) return;
  int ns = mapping[src[e]];
  int nd = mapping[dst[e]];
  float m = (emask ? emask[e] : 1.0f) * ((ns >= 0 && nd >= 0) ? 1.0f : 0.0f);
  nsrc[e] = ns > 0 ? ns : 0;
  ndst[e] = nd > 0 ? nd : 0;
  nmask[e] = m;
}

// ---------------- head ----------------
__global__ void meanpool(const float* __restrict__ h, float* __restrict__ pooled, int n, int F) {
  int f = blockIdx.x * blockDim.x + threadIdx.x;
  if (f >= F) return;
  float s = 0.0f;
  for (int i = 0; i < n; ++i) s += h[(size_t)i * F + f];
  pooled[f] = s / (float)n;
}

__global__ void head_kernel(const float* __restrict__ pooled, const float* __restrict__ W,
                            const float* __restrict__ b, float* __restrict__ out, int F, int O) {
  __shared__ float logit[32];
  int tid = threadIdx.x;
  if (tid < O) {
    float s = b[tid];
    for (int f = 0; f < F; ++f) s += pooled[f] * W[(size_t)f * O + tid];
    logit[tid] = s;
  }
  __syncthreads();
  if (tid == 0) {
    float m = logit[0];
    for (int o = 1; o < O; ++o) m = fmaxf(m, logit[o]);
    float se = 0.0f;
    for (int o = 0; o < O; ++o) se += expf(logit[o] - m);
    float lse = m + logf(se);
    for (int o = 0; o < O; ++o) out[o] = logit[o] - lse;
  }
}

// ---------------- host orchestration ----------------
static inline int cdiv(long a, int b) { return (int)((a + b - 1) / b); }
static inline int ilog2(int v) { int s = 0; while ((1 << s) < v) ++s; return s; }

extern "C" void kernel_launch(void* const* d_in, const int* in_sizes, int n_in,
                              void* d_out, int out_size, void* d_ws, size_t ws_size,
                              hipStream_t stream) {
  const float* x   = (const float*)d_in[0];
  const int*   ei  = (const int*)d_in[1];
  const int Nn = in_sizes[0] / 128;   // 20000
  const int E  = in_sizes[1] / 2;     // 160000
  const int* src0 = ei;
  const int* dst0 = ei + E;
  const float *W[6], *bb[6];
  for (int i = 0; i < 6; ++i) { W[i] = (const float*)d_in[2 + 2 * i]; bb[i] = (const float*)d_in[3 + 2 * i]; }
  const float* p1   = (const float*)d_in[14];
  const float* p2   = (const float*)d_in[15];
  const float* Wlin = (const float*)d_in[16];
  const float* blin = (const float*)d_in[17];
  float* out = (float*)d_out;

  const int K1 = (9 * Nn + 9) / 10;   // 18000
  const int K2 = (9 * K1 + 9) / 10;   // 16200

  // workspace carve
  char* wp = (char*)d_ws;
  auto carve = [&](size_t bytes) -> void* {
    void* p = wp; wp += (bytes + 255) & ~(size_t)255; return p;
  };
  const size_t big = (size_t)Nn * 1024 * sizeof(float);
  float* buf0   = (float*)carve(big);
  float* buf1   = (float*)carve(big);
  float* buf2   = (float*)carve(big);
  float* deg    = (float*)carve((size_t)Nn * 4);
  float* nrm    = (float*)carve((size_t)E * 4);
  float* score  = (float*)carve((size_t)Nn * 4);
  int*   mapping= (int*)  carve((size_t)Nn * 4);
  int*   srcA   = (int*)  carve((size_t)E * 4);
  int*   dstA   = (int*)  carve((size_t)E * 4);
  float* emaskA = (float*)carve((size_t)E * 4);
  int*   perm1  = (int*)  carve((size_t)K1 * 4);
  int*   perm2  = (int*)  carve((size_t)K2 * 4);
  float* pnorm  = (float*)carve(256);
  float* pooled = (float*)carve(1024 * 4);

  auto compute_norm = [&](int n, const int* esrc, const int* edst, const float* em) {
    fill_f32<<<cdiv(n, 256), 256, 0, stream>>>(deg, 1.0f, n);
    deg_accum<<<cdiv(E, 256), 256, 0, stream>>>(edst, em, deg, E);
    edge_norm<<<cdiv(E, 256), 256, 0, stream>>>(esrc, edst, em, deg, nrm, E);
  };

  auto run_gcn = [&](const float* in, int n, int fin, int fout,
                     const float* Wm, const float* bv,
                     const int* esrc, const int* edst,
                     float* hbuf, float* abuf) {
    int totalWaves = (n / 16) * (fout / 64);
    gemm_wmma<<<cdiv(totalWaves, 8), 256, 0, stream>>>(in, Wm, hbuf, n, fout, fin, totalWaves);
    int nf = n * fout;
    int fs = ilog2(fout);
    fill_f32<<<cdiv(nf, 256), 256, 0, stream>>>(abuf, 0.0f, nf);
    long tot = (long)E * (fout / 4);
    agg_edges<<<cdiv(tot, 256), 256, 0, stream>>>(esrc, edst, nrm, hbuf, abuf, E, fout, fs - 2);
    gcn_finalize<<<cdiv(nf, 256), 256, 0, stream>>>(abuf, hbuf, deg, bv, abuf, n, fout, fs);
  };

  auto run_pool = [&](const float* xin, int n, int Fdim, const float* pvec, int k,
                      int* permBuf, float* dout_perm, float* xout) {
    pnorm_kernel<<<1, 256, 0, stream>>>(pvec, Fdim, pnorm);
    score_kernel<<<cdiv((long)n * 32, 256), 256, 0, stream>>>(xin, pvec, pnorm, score, n, Fdim);
    int total = 1;
    while (total < n) total <<= 1;                    // 32768
    size_t lds = (size_t)total * sizeof(u64);         // 256 KB fits 320 KB LDS
    hipFuncSetAttribute((const void*)topk_sort, hipFuncAttributeMaxDynamicSharedMemorySize, (int)lds);
    topk_sort<<<1, 1024, lds, stream>>>(score, n, total, k, permBuf, dout_perm);
    gather_scale<<<cdiv((long)k * Fdim, 256), 256, 0, stream>>>(xin, permBuf, score, xout, k, Fdim, ilog2(Fdim));
  };

  // ---- stage 1: 3 GCNs on the full graph ----
  compute_norm(Nn, src0, dst0, nullptr);
  run_gcn(x,    Nn, 128,  64, W[0], bb[0], src0, dst0, buf0, buf1);  // -> buf1
  run_gcn(buf1, Nn,  64, 128, W[1], bb[1], src0, dst0, buf2, buf0);  // -> buf0
  run_gcn(buf0, Nn, 128, 256, W[2], bb[2], src0, dst0, buf1, buf2);  // -> buf2

  // ---- pool 1 ----
  run_pool(buf2, Nn, 256, p1, K1, perm1, out + 10, buf0);            // xk -> buf0
  fill_i32<<<cdiv(Nn, 256), 256, 0, stream>>>(mapping, -1, Nn);
  scatter_map<<<cdiv(K1, 256), 256, 0, stream>>>(perm1, mapping, K1);
  relabel<<<cdiv(E, 256), 256, 0, stream>>>(src0, dst0, nullptr, mapping, srcA, dstA, emaskA, E);

  // ---- stage 2: 3 GCNs on the pooled graph ----
  compute_norm(K1, srcA, dstA, emaskA);
  run_gcn(buf0, K1, 256,  256, W[3], bb[3], srcA, dstA, buf1, buf2); // -> buf2
  run_gcn(buf2, K1, 256,  512, W[4], bb[4], srcA, dstA, buf0, buf1); // -> buf1
  run_gcn(buf1, K1, 512, 1024, W[5], bb[5], srcA, dstA, buf2, buf0); // -> buf0

  // ---- pool 2 (edges unused afterwards) ----
  run_pool(buf0, K1, 1024, p2, K2, perm2, out + 10 + K1, buf1);      // xk -> buf1

  // ---- head ----
  meanpool<<<cdiv(1024, 256), 256, 0, stream>>>(buf1, pooled, K2, 1024);
  head_kernel<<<1, 32, 0, stream>>>(pooled, Wlin, blin, out, 1024, 10);
}